// DeltaRuleMemory_16071767622135
// MI455X (gfx1250) — compile-verified
//
#include <hip/hip_runtime.h>
#include <math.h>

typedef float v2f __attribute__((ext_vector_type(2)));
typedef float v8f __attribute__((ext_vector_type(8)));

#define B_   16
#define S_   4096
#define DK_  256
#define DV_  256
#define CHK  16
#define NCHK (S_ / CHK)
#define SPLIT 8
#define VROWS (DV_ / SPLIT)       // 32 rows of M per block
#define NTHREADS 256
#define NWAVES (NTHREADS / 32)    // 8 wave32 waves
#define MS 260                     // padded row stride (even -> b64-mergeable, conflict-free)
#define US 33                      // padded row stride for P0T
#define UT 18                      // padded row stride for Uv[v][t] (even -> b64 A-fragments)
#define VS 34                      // padded row stride for Vc[t][v]

#define LR_C 0.1f
#define A_C  1.1f                  // 1 + LR
#define EPS_C 1e-6f

__global__ __launch_bounds__(NTHREADS)
void DeltaRuleMemory_chunked_wmma(const float* __restrict__ mem,
                                  const float* __restrict__ key,
                                  const float* __restrict__ val,
                                  float* __restrict__ out) {
    extern __shared__ float smem[];
    float* Msub = smem;                        // VROWS x MS
    float* Kn   = Msub + VROWS * MS;           // CHK x MS   (normalized keys, row t)
    float* P0Ta = Kn   + CHK * MS;             // CHK x US   (P0^T partial, K-half 0)
    float* P0Tb = P0Ta + CHK * US;             // CHK x US   (P0^T partial, K-half 1)
    float* Uv   = P0Tb + CHK * US;             // VROWS x UT ([v][t], transposed U)
    float* Ga   = Uv   + VROWS * UT;           // 16x16 gram partial, K-half 0
    float* Gb   = Ga   + 256;                  // 16x16 gram partial, K-half 1
    float* Vc   = Gb   + 256;                  // CHK x VS   (staged value slice)
    float* red  = Vc   + CHK * VS;             // 16x16 norm partials
    float* inv  = red  + 256;                  // 16 inverse norms

    const int tid  = threadIdx.x;
    const int wave = tid >> 5;
    const int lane = tid & 31;
    const int hi   = lane >> 4;                // wave32 half select
    const int lm   = lane & 15;

    const int blk = blockIdx.x;
    const int b   = blk / SPLIT;
    const int v0  = (blk % SPLIT) * VROWS;

    const float* memB = mem + (size_t)b * DV_ * DK_ + (size_t)v0 * DK_;
    const float* keyB = key + (size_t)b * S_ * DK_;
    const float* valB = val + (size_t)b * S_ * DV_;
    float*       outB = out + (size_t)b * DV_ * DK_ + (size_t)v0 * DK_;

    // ---- load resident M slice into LDS ----
    for (int i = tid; i < VROWS * DK_; i += NTHREADS) {
        int r = i >> 8, k = i & 255;
        Msub[r * MS + k] = memB[r * DK_ + k];
    }
    __syncthreads();

    const int tA = tid >> 4;                   // chunk row 0..15
    const int sg = tid & 15;
    const int d0 = sg * 16;
    const int jV = sg * 2;                     // value-slice column pair

    for (int c = 0; c < NCHK; ++c) {
        // ================= Phase A: stage keys (normalize) + values =================
        float4 rv[4];
        float ss = 0.f;
        const float* kr = keyB + (size_t)(c * CHK + tA) * DK_ + d0;
        #pragma unroll
        for (int j = 0; j < 4; ++j) {
            rv[j] = ((const float4*)kr)[j];
            ss += rv[j].x * rv[j].x + rv[j].y * rv[j].y +
                  rv[j].z * rv[j].z + rv[j].w * rv[j].w;
        }
        // stage this chunk's value slice into LDS (coalesced; hides latency
        // behind the Phase-B WMMA work so the serial solve reads LDS only)
        {
            const float* vg = valB + (size_t)(c * CHK + tA) * DV_ + v0 + jV;
            Vc[tA * VS + jV]     = vg[0];
            Vc[tA * VS + jV + 1] = vg[1];
        }
        if (c + 1 < NCHK)   // hint next chunk's keys toward the caches
            __builtin_prefetch(keyB + (size_t)((c + 1) * CHK + tA) * DK_ + d0, 0, 0);
        red[tA * 16 + sg] = ss;
        __syncthreads();
        if (tid < 16) {
            float s = 0.f;
            #pragma unroll
            for (int j = 0; j < 16; ++j) s += red[tid * 16 + j];
            inv[tid] = 1.f / (sqrtf(s) + EPS_C);
        }
        __syncthreads();
        {
            float sc = inv[tA];
            float* kw = Kn + tA * MS + d0;
            #pragma unroll
            for (int j = 0; j < 4; ++j) {
                kw[4 * j + 0] = rv[j].x * sc;
                kw[4 * j + 1] = rv[j].y * sc;
                kw[4 * j + 2] = rv[j].z * sc;
                kw[4 * j + 3] = rv[j].w * sc;
            }
        }
        __syncthreads();

        // ===== Phase B: [P0 ; G] = [Msub ; Kn] * Kn^T, K split across wave pairs =====
        if (wave < 6) {
            const int tile = wave >> 1;        // 0,1: P0 tiles; 2: Gram tile
            const int kh   = wave & 1;         // K half
            const float* arow = (tile < 2 ? Msub + (tile * 16 + lm) * MS
                                          : Kn + lm * MS) + 2 * hi;
            const float* brow = Kn + lm * MS + 2 * hi;
            v8f acc0 = {}, acc1 = {};
            const int kbeg = kh * 128;
            for (int k0 = kbeg; k0 < kbeg + 128; k0 += 8) {
                v2f a0 = *(const v2f*)(arow + k0);
                v2f b0 = *(const v2f*)(brow + k0);
                acc0 = __builtin_amdgcn_wmma_f32_16x16x4_f32(
                           false, a0, false, b0, (short)0, acc0, false, false);
                v2f a1 = *(const v2f*)(arow + k0 + 4);
                v2f b1 = *(const v2f*)(brow + k0 + 4);
                acc1 = __builtin_amdgcn_wmma_f32_16x16x4_f32(
                           false, a1, false, b1, (short)0, acc1, false, false);
            }
            v8f acc = acc0 + acc1;
            if (tile < 2) {                    // store transposed: P0T[t][v]
                float* dst = kh ? P0Tb : P0Ta;
                #pragma unroll
                for (int r = 0; r < 8; ++r)
                    dst[lm * US + tile * 16 + r + 8 * hi] = acc[r];
            } else {                           // gram G[t][s]
                float* dst = kh ? Gb : Ga;
                #pragma unroll
                for (int r = 0; r < 8; ++r)
                    dst[(r + 8 * hi) * 16 + lm] = acc[r];
            }
        }
        __syncthreads();

        // ===== Phase C: per-component triangular solve (no cross-thread deps) =====
        if (tid < VROWS) {
            const int v = tid;
            float u[CHK];
            #pragma unroll
            for (int t = 0; t < CHK; ++t) {
                float x = LR_C * Vc[t * VS + v]
                        - A_C * (P0Ta[t * US + v] + P0Tb[t * US + v]);
                for (int s2 = 0; s2 < t; ++s2)
                    x -= A_C * (Ga[t * 16 + s2] + Gb[t * 16 + s2]) * u[s2];
                u[t] = x;
            }
            float* uw = Uv + v * UT;           // contiguous -> vectorizable stores
            #pragma unroll
            for (int t = 0; t < CHK; ++t) uw[t] = u[t];
        }
        __syncthreads();

        // ===== Phase D: Msub += U^T * Kn  (VROWS x 256, K = 16) =====
        #pragma unroll
        for (int i = 0; i < (VROWS / 16) * (DK_ / 16) / NWAVES; ++i) {
            int tl = wave + i * NWAVES;
            int vt = tl & (VROWS / 16 - 1);
            int kt = tl / (VROWS / 16);
            float* crow = Msub + (vt * 16) * MS + kt * 16 + lm;
            v8f acc;
            #pragma unroll
            for (int r = 0; r < 8; ++r) acc[r] = crow[(r + 8 * hi) * MS];
            const float* urow = Uv + (vt * 16 + lm) * UT + 2 * hi;
            #pragma unroll
            for (int k0 = 0; k0 < CHK; k0 += 4) {
                v2f a = *(const v2f*)(urow + k0);      // b64: A[m][k]=Uv[v][t]
                v2f bb; bb.x = Kn[(k0 + 2 * hi) * MS + kt * 16 + lm];
                        bb.y = Kn[(k0 + 2 * hi + 1) * MS + kt * 16 + lm];
                acc = __builtin_amdgcn_wmma_f32_16x16x4_f32(
                          false, a, false, bb, (short)0, acc, false, false);
            }
            #pragma unroll
            for (int r = 0; r < 8; ++r) crow[(r + 8 * hi) * MS] = acc[r];
        }
        __syncthreads();
    }

    // ---- write final M slice ----
    for (int i = tid; i < VROWS * DK_; i += NTHREADS) {
        int r = i >> 8, k = i & 255;
        outB[r * DK_ + k] = Msub[r * MS + k];
    }
}

extern "C" void kernel_launch(void* const* d_in, const int* in_sizes, int n_in,
                              void* d_out, int out_size, void* d_ws, size_t ws_size,
                              hipStream_t stream) {
    (void)in_sizes; (void)n_in; (void)out_size; (void)d_ws; (void)ws_size;
    const float* mem = (const float*)d_in[0];
    const float* key = (const float*)d_in[1];
    const float* val = (const float*)d_in[2];
    float* out = (float*)d_out;

    size_t smemFloats = (size_t)(VROWS * MS)   // Msub
                      + CHK * MS               // Kn
                      + 2 * CHK * US           // P0Ta, P0Tb
                      + VROWS * UT             // Uv
                      + 2 * 256                // Ga, Gb
                      + CHK * VS               // Vc
                      + 256 + 16;              // red, inv
    size_t smemBytes = smemFloats * sizeof(float);   // ~62 KB, fits 320 KB WGP LDS

    dim3 grid(B_ * SPLIT);        // 128 workgroups: 16 batches x 8 DV-slices
    dim3 block(NTHREADS);         // 8 wave32 waves
    hipLaunchKernelGGL(DeltaRuleMemory_chunked_wmma, grid, block, smemBytes, stream,
                       mem, key, val, out);
}